// DenoisingBilateral_52785148068075
// MI455X (gfx1250) — compile-verified
//
#include <hip/hip_runtime.h>

typedef __attribute__((ext_vector_type(2))) float v2f;
typedef __attribute__((ext_vector_type(8))) float v8f;

#define B_IMG 2
#define H_IMG 512
#define W_IMG 512
#define DEPTH 32
#define HW_IMG (H_IMG * W_IMG)
#define NPIX (B_IMG * HW_IMG)              // 524288
#define NGRID ((size_t)NPIX * DEPTH)       // 16777216
#define BANDW 80                           // max band width (pad<=32 -> 16+2*32)

static __device__ __forceinline__ float clampf(float x, float a, float b) {
  return fminf(fmaxf(x, a), b);
}

// ---------------- RGB -> YUV planes ----------------
__global__ void k_rgb2yuv(const float* __restrict__ img,
                          float* __restrict__ yP,
                          float* __restrict__ uP,
                          float* __restrict__ vP) {
  int i = blockIdx.x * blockDim.x + threadIdx.x;
  if (i >= NPIX) return;
  int b = i / HW_IMG;
  int p = i - b * HW_IMG;
  const float* base = img + (size_t)b * 3 * HW_IMG + p;
  float r  = base[0];
  float g  = base[HW_IMG];
  float bl = base[2 * HW_IMG];
  yP[i] =  0.299f   * r + 0.587f   * g + 0.114f   * bl;
  uP[i] = -0.14713f * r - 0.28886f * g + 0.436f   * bl;
  vP[i] =  0.615f   * r - 0.51499f * g - 0.10001f * bl;
}

// ---------------- guide -> (z0, wz) ----------------
__global__ void k_ziwz(const float* __restrict__ guide,
                       int* __restrict__ zi, float* __restrict__ wz) {
  int i = blockIdx.x * blockDim.x + threadIdx.x;
  if (i >= NPIX) return;
  float z = clampf(guide[i], 0.0f, 1.0f) * (float)(DEPTH - 1);
  int z0 = (int)floorf(z);
  z0 = min(max(z0, 0), DEPTH - 2);
  zi[i] = z0;
  wz[i] = z - (float)z0;
}

// ---------------- splat one channel into grid[pix][32] ----------------
__global__ void k_splat(const float* __restrict__ val,   // nullptr => ones
                        const int* __restrict__ zi, const float* __restrict__ wz,
                        float* __restrict__ grid) {
  int i = blockIdx.x * blockDim.x + threadIdx.x;
  if (i >= NPIX) return;
  float v  = val ? val[i] : 1.0f;
  int   z0 = zi[i];
  float w  = wz[i];
  float v0 = v * (1.0f - w);
  float v1 = v * w;
  float* gp = grid + (size_t)i * DEPTH;
#pragma unroll
  for (int zb = 0; zb < DEPTH; zb += 4) {
    float4 o;
    o.x = (zb + 0 == z0) ? v0 : ((zb + 0 == z0 + 1) ? v1 : 0.0f);
    o.y = (zb + 1 == z0) ? v0 : ((zb + 1 == z0 + 1) ? v1 : 0.0f);
    o.z = (zb + 2 == z0) ? v0 : ((zb + 2 == z0 + 1) ? v1 : 0.0f);
    o.w = (zb + 3 == z0) ? v0 : ((zb + 3 == z0 + 1) ? v1 : 0.0f);
    *(float4*)(gp + zb) = o;
  }
}

// =====================================================================
// Unified spatial blur along H (stride=W) or W (stride=1) as a banded
// matmul on WMMA, with the input tile staged LDS via async copies.
//
// Block: one line (fixed b,h for W-axis / fixed b,w for H-axis),
// 64 output positions. Tile in LDS: (64+2*pad) positions x 32 z (f32),
// contiguous-per-position. Band table in LDS: band[m][k] = filt[k-m-pad+r].
// Wave wv (0..7): subtile wsub=(wv&3)*16 positions, zt=(wv>>2) z-half.
//   D(16 pos x 16 z) = sum_kb A(16x4 of band) * B(4x16 of tile)
// =====================================================================
__global__ void __launch_bounds__(256)
k_sblur_wmma(const float* __restrict__ gin, float* __restrict__ gout,
             const float* __restrict__ filt, int ntaps, int axis) {
  __shared__ float tileLds[128 * DEPTH];       // up to 16 KB
  __shared__ float bandLds[16 * BANDW];        // 5 KB

  const int r    = (ntaps - 1) >> 1;
  const int pad  = (r + 3) & ~3;               // 8 for r=6, 32 for r=30
  const int nkb  = 4 + (pad >> 1);             // k-blocks of 4: 8 or 20
  const int nld  = 64 + 2 * pad;               // loaded positions: 80 or 128

  // Block -> line decode
  const int outer = blockIdx.x >> 3;           // b*H+h (axis 0) or b*W+w (axis 1)
  const int p0    = (blockIdx.x & 7) * 64;     // first output position on the line
  size_t base;  int stride;
  if (axis == 0) { base = (size_t)outer * W_IMG;                   stride = 1; }
  else           { base = (size_t)(outer >> 9) * HW_IMG + (outer & 511); stride = W_IMG; }

  // ---- band table: band[m][k] = filt[k - m - pad + r] (zero outside) ----
  for (int idx = threadIdx.x; idx < 16 * BANDW; idx += 256) {
    int m = idx / BANDW, k = idx - m * BANDW;
    int d = k - m - pad + r;
    bandLds[idx] = (d >= 0 && d < ntaps) ? filt[d] : 0.0f;
  }

  // ---- stage tile: positions p0-pad .. p0+63+pad, zero outside image ----
  const unsigned tileBase = (unsigned)(uintptr_t)&tileLds[0];   // LDS byte offset
  for (int slot = threadIdx.x; slot < nld * 8; slot += 256) {   // float4 slots
    int pl = slot >> 3;                        // local position
    int zq = slot & 7;                         // z quad
    int sp = p0 - pad + pl;                    // source position on line
    if (sp >= 0 && sp < 512) {
      const float* ga = gin + ((size_t)(base + (size_t)sp * stride)) * DEPTH + zq * 4;
      unsigned la = tileBase + (unsigned)slot * 16;
      asm volatile("global_load_async_to_lds_b128 %0, %1, off"
                   :: "v"(la), "v"(ga) : "memory");
    } else {
      *(float4*)&tileLds[slot * 4] = make_float4(0.f, 0.f, 0.f, 0.f);
    }
  }
  asm volatile("s_wait_asynccnt 0" ::: "memory");
  __syncthreads();

  // ---- compute: one 16pos x 16z tile per wave ----
  const int lane = threadIdx.x & 31;
  const int wv   = threadIdx.x >> 5;           // 0..7
  const int wsub = (wv & 3) * 16;              // position subtile
  const int zt   = wv >> 2;                    // z half (0/1)
  const int n    = lane & 15;                  // column (z) / band row (m)
  const int kh   = lane >> 4;                  // K half

  v8f acc = {};
  for (int kb = 0; kb < nkb; ++kb) {
    const int k0 = kb * 4 + 2 * kh;
    const int lw = wsub + k0;                  // local tile position of B[k0][*]
    v2f bt;
    bt.x = tileLds[lw * DEPTH + zt * 16 + n];
    bt.y = tileLds[(lw + 1) * DEPTH + zt * 16 + n];
    v2f a;
    a.x = bandLds[n * BANDW + k0];
    a.y = bandLds[n * BANDW + k0 + 1];
    acc = __builtin_amdgcn_wmma_f32_16x16x4_f32(false, a, false, bt,
                                                (short)0, acc, false, false);
  }

  // ---- store D: rows m = 8*kh + v (output positions), col n (z) ----
#pragma unroll
  for (int v = 0; v < 8; ++v) {
    int m   = 8 * kh + v;
    int pos = p0 + wsub + m;
    gout[((size_t)(base + (size_t)pos * stride)) * DEPTH + zt * 16 + n] = acc[v];
  }
}

// =====================================================================
// z-blur: banded 32x32 matmul via WMMA f32 16x16x4 (compile-time band)
// =====================================================================
template <int R>
__global__ void __launch_bounds__(256)
k_zblur_wmma(const float* __restrict__ gin, float* __restrict__ gout,
             const float* __restrict__ filt) {
  constexpr int NT = 2 * R + 1;
  __shared__ float kz[DEPTH][DEPTH];
  for (int idx = threadIdx.x; idx < DEPTH * DEPTH; idx += blockDim.x) {
    int zo = idx >> 5, zn = idx & 31;
    int d = zn - zo + R;
    kz[zo][zn] = (d >= 0 && d < NT) ? filt[d] : 0.0f;
  }
  __syncthreads();

  const int lane = threadIdx.x & 31;
  const int wave = blockIdx.x * (blockDim.x >> 5) + (threadIdx.x >> 5);
  const int n    = lane & 15;
  const int kh   = lane >> 4;
  const size_t gbase = ((size_t)(wave * 16 + n)) * DEPTH;
  const float* gp = gin + gbase;

  v8f acc0 = {};
  v8f acc1 = {};
#pragma unroll
  for (int kb = 0; kb < 8; ++kb) {
    const int k0 = kb * 4 + 2 * kh;
    v2f bt;
    bt.x = gp[k0];
    bt.y = gp[k0 + 1];
    if (kb * 4 <= 15 + R) {                   // compile-time: band hits M-tile 0
      v2f a; a.x = kz[n][k0];      a.y = kz[n][k0 + 1];
      acc0 = __builtin_amdgcn_wmma_f32_16x16x4_f32(false, a, false, bt,
                                                   (short)0, acc0, false, false);
    }
    if (kb * 4 + 3 >= 16 - R) {               // compile-time: band hits M-tile 1
      v2f a; a.x = kz[16 + n][k0]; a.y = kz[16 + n][k0 + 1];
      acc1 = __builtin_amdgcn_wmma_f32_16x16x4_f32(false, a, false, bt,
                                                   (short)0, acc1, false, false);
    }
  }
  float* op = gout + gbase + kh * 8;
  *(v8f*)op        = acc0;
  *(v8f*)(op + 16) = acc1;
}

// Dynamic-radius fallback (same algorithm, runtime band tests)
__global__ void __launch_bounds__(256)
k_zblur_wmma_dyn(const float* __restrict__ gin, float* __restrict__ gout,
                 const float* __restrict__ filt, int ntaps) {
  __shared__ float kz[DEPTH][DEPTH];
  const int r = (ntaps - 1) >> 1;
  for (int idx = threadIdx.x; idx < DEPTH * DEPTH; idx += blockDim.x) {
    int zo = idx >> 5, zn = idx & 31;
    int d = zn - zo + r;
    kz[zo][zn] = (d >= 0 && d < ntaps) ? filt[d] : 0.0f;
  }
  __syncthreads();

  const int lane = threadIdx.x & 31;
  const int wave = blockIdx.x * (blockDim.x >> 5) + (threadIdx.x >> 5);
  const int n    = lane & 15;
  const int kh   = lane >> 4;
  const size_t gbase = ((size_t)(wave * 16 + n)) * DEPTH;
  const float* gp = gin + gbase;

  v8f acc0 = {};
  v8f acc1 = {};
#pragma unroll
  for (int kb = 0; kb < 8; ++kb) {
    const int k0 = kb * 4 + 2 * kh;
    v2f bt;
    bt.x = gp[k0];
    bt.y = gp[k0 + 1];
    if (kb * 4 <= 15 + r) {
      v2f a; a.x = kz[n][k0];      a.y = kz[n][k0 + 1];
      acc0 = __builtin_amdgcn_wmma_f32_16x16x4_f32(false, a, false, bt,
                                                   (short)0, acc0, false, false);
    }
    if (kb * 4 + 3 >= 16 - r) {
      v2f a; a.x = kz[16 + n][k0]; a.y = kz[16 + n][k0 + 1];
      acc1 = __builtin_amdgcn_wmma_f32_16x16x4_f32(false, a, false, bt,
                                                   (short)0, acc1, false, false);
    }
  }
  float* op = gout + gbase + kh * 8;
  *(v8f*)op        = acc0;
  *(v8f*)(op + 16) = acc1;
}

// ---------------- slice: linear interp in z at full-res (x,y) ----------------
__global__ void k_slice(const float* __restrict__ grid,
                        const int* __restrict__ zi, const float* __restrict__ wz,
                        float* __restrict__ out) {
  int i = blockIdx.x * blockDim.x + threadIdx.x;
  if (i >= NPIX) return;
  int z0 = zi[i];
  float w = wz[i];
  const float* gp = grid + (size_t)i * DEPTH;
  out[i] = gp[z0] * (1.0f - w) + gp[z0 + 1] * w;
}

__global__ void k_lumfinal(const float* __restrict__ slY, const float* __restrict__ slW,
                           float* __restrict__ outLum) {
  int i = blockIdx.x * blockDim.x + threadIdx.x;
  if (i >= NPIX) return;
  outLum[i] = slY[i] / fmaxf(slW[i], 1e-8f);
}

// ---------------- combine + YUV -> RGB ----------------
__global__ void k_final(const float* __restrict__ outLum,
                        const float* __restrict__ slU, const float* __restrict__ slV,
                        const float* __restrict__ slW, float* __restrict__ out) {
  int i = blockIdx.x * blockDim.x + threadIdx.x;
  if (i >= NPIX) return;
  int b = i / HW_IMG;
  int p = i - b * HW_IMG;
  float y  = outLum[i];
  float wd = fmaxf(slW[i], 1e-8f);
  float u  = slU[i] / wd;
  float v  = slV[i] / wd;
  float R = y + 1.1402509f * v;
  float G = y - 0.3947313f * u - 0.5808091f * v;
  float B = y + 2.0325203f * u;
  float* o = out + (size_t)b * 3 * HW_IMG + p;
  o[0]          = R;
  o[HW_IMG]     = G;
  o[2 * HW_IMG] = B;
}

extern "C" void kernel_launch(void* const* d_in, const int* in_sizes, int n_in,
                              void* d_out, int out_size, void* d_ws, size_t ws_size,
                              hipStream_t stream) {
  (void)n_in; (void)out_size; (void)ws_size;
  const float* image = (const float*)d_in[0];
  const float* f_s   = (const float*)d_in[1]; int n_s  = in_sizes[1];  // 13
  const float* f_r   = (const float*)d_in[2]; int n_r  = in_sizes[2];  // 13
  const float* f_sc  = (const float*)d_in[3]; int n_sc = in_sizes[3];  // 61
  const float* f_rc  = (const float*)d_in[4]; int n_rc = in_sizes[4];  // 13
  float* out = (float*)d_out;

  // Workspace carve-up (~153 MB of f32)
  float* ws     = (float*)d_ws;
  float* yP     = ws; ws += NPIX;
  float* uP     = ws; ws += NPIX;
  float* vP     = ws; ws += NPIX;
  float* gA     = ws; ws += NGRID;
  float* gB     = ws; ws += NGRID;
  float* wz     = ws; ws += NPIX;
  int*   zi     = (int*)ws; ws += NPIX;
  float* slA    = ws; ws += NPIX;
  float* slB    = ws; ws += NPIX;
  float* slW    = ws; ws += NPIX;
  float* outLum = ws; ws += NPIX;

  const dim3 blk(256);
  const int gPix = (NPIX + 255) / 256;
  const int gS   = B_IMG * 512 * (512 / 64);   // 8192 blocks, both axes
  const int gZ   = NPIX / (16 * 8);            // 4096 blocks, 8 full waves each

  k_rgb2yuv<<<gPix, blk, 0, stream>>>(image, yP, uP, vP);

  auto zblur = [&](const float* gi, float* go, const float* fr, int nr) {
    if (nr == 13) k_zblur_wmma<6><<<gZ, blk, 0, stream>>>(gi, go, fr);
    else          k_zblur_wmma_dyn<<<gZ, blk, 0, stream>>>(gi, go, fr, nr);
  };

  auto pipeline = [&](const float* val, const float* fs, int ns,
                      const float* fr, int nr, float* sliced) {
    k_splat<<<gPix, blk, 0, stream>>>(val, zi, wz, gA);
    k_sblur_wmma<<<gS, blk, 0, stream>>>(gA, gB, fs, ns, /*axis=H*/1);
    k_sblur_wmma<<<gS, blk, 0, stream>>>(gB, gA, fs, ns, /*axis=W*/0);
    zblur(gA, gB, fr, nr);
    k_slice<<<gPix, blk, 0, stream>>>(gB, zi, wz, sliced);
  };

  // ---- Luma bilateral grid (guide = clip(Y)) ----
  k_ziwz<<<gPix, blk, 0, stream>>>(yP, zi, wz);
  pipeline(yP,      f_s, n_s, f_r, n_r, slA);   // value channel
  pipeline(nullptr, f_s, n_s, f_r, n_r, slW);   // homogeneous weight channel
  k_lumfinal<<<gPix, blk, 0, stream>>>(slA, slW, outLum);

  // ---- Chroma bilateral grid (guide = out_lum) ----
  k_ziwz<<<gPix, blk, 0, stream>>>(outLum, zi, wz);
  pipeline(uP,      f_sc, n_sc, f_rc, n_rc, slA);
  pipeline(vP,      f_sc, n_sc, f_rc, n_rc, slB);
  pipeline(nullptr, f_sc, n_sc, f_rc, n_rc, slW);

  k_final<<<gPix, blk, 0, stream>>>(outLum, slA, slB, slW, out);
}